// LimitedAdaptiveComb1d_46170898432118
// MI455X (gfx1250) — compile-verified
//
#include <hip/hip_runtime.h>
#include <hip/hip_bf16.h>

typedef __attribute__((ext_vector_type(2))) float v2f;
typedef __attribute__((ext_vector_type(8))) float v8f;

#define FD      128
#define KT      15        // comb taps
#define FS      160
#define OV      40
#define OUTL    (FS + OV) // 200
#define NF      500
#define NB      32
#define NSAMP   (NF * FS) // 80000
#define PSTRIDE 20        // 15 kernels + gain + ggain, padded

#define LOG_GAIN_LIMIT 1.1512925464970229f   // 10*C
#define GAIN_A         0.6907755278982137f   // 6*C  (GAIN_B == 0)
#define PI_F           3.14159265358979323846f

__device__ __forceinline__ float xat(const float* __restrict__ xb, int idx) {
    return (idx >= 0 && idx < NSAMP) ? xb[idx] : 0.0f;
}

// ---------------------------------------------------------------------------
// Kernel 1: per-frame params via fp32 WMMA.
// One wave32 per tile of 16 frames.  A = features[16 x 4-chunk], B = weights.
// B operands both live in one zero-padded 32x128 LDS array so the inner loop
// is pure ds_load_b64 + v_wmma (no exec-masked selects):
//   rows  0..14 : ck_w          -> acc1 columns 0..14 (comb kernel taps)
//   row     15  : fg_w          -> acc1 column  15    (frame-gain proj)
//   row     16  : gg_w          -> acc2 column   0    (global-gain proj)
//   rows 17..31 : zeros         -> acc2 columns 1..15 (unused)
// ---------------------------------------------------------------------------
__global__ void __launch_bounds__(32)
frame_params_wmma(const float* __restrict__ features,  // (B*NF, FD)
                  const float* __restrict__ ck_w,      // (K, FD)
                  const float* __restrict__ ck_b,      // (K)
                  const float* __restrict__ fg_w,      // (FD)
                  const float* __restrict__ fg_b,      // (1)
                  const float* __restrict__ gg_w,      // (FD)
                  const float* __restrict__ gg_b,      // (1)
                  float* __restrict__ params)          // (B*NF, PSTRIDE)
{
    const int lane = threadIdx.x;      // 0..31
    const int half = lane >> 4;        // K-split half
    const int col  = lane & 15;        // N (B/C/D column), also M for A rows

    const int F0 = blockIdx.x * 16;    // 16000 frames -> exactly 1000 tiles

    __shared__ float sW[32 * FD];      // see layout above (16 KB)
    __shared__ float smat[16][17];     // result staging: [row][0..15]=acc1, [16]=gg

    for (int i = lane; i < 32 * FD; i += 32) {
        const int r = i >> 7, d = i & (FD - 1);
        float v = 0.0f;
        if (r < KT)       v = ck_w[r * FD + d];
        else if (r == KT) v = fg_w[d];
        else if (r == 16) v = gg_w[d];
        sW[i] = v;
    }
    __syncthreads();

    const float* __restrict__ arow = features + (size_t)(F0 + col) * FD;
    const float* __restrict__ b1p  = sW + col * FD;          // W1 column
    const float* __restrict__ b2p  = sW + (16 + col) * FD;   // W2 column (zero-padded)

    v8f acc1 = {0.f, 0.f, 0.f, 0.f, 0.f, 0.f, 0.f, 0.f};
    v8f acc2 = {0.f, 0.f, 0.f, 0.f, 0.f, 0.f, 0.f, 0.f};

    for (int d0 = 0; d0 < FD; d0 += 4) {
        const int d = d0 + 2 * half;   // VGPR0 holds K=d, VGPR1 holds K=d+1
        v2f a;  a.x  = arow[d];  a.y  = arow[d + 1];
        v2f b1; b1.x = b1p[d];   b1.y = b1p[d + 1];
        v2f b2; b2.x = b2p[d];   b2.y = b2p[d + 1];
        acc1 = __builtin_amdgcn_wmma_f32_16x16x4_f32(false, a, false, b1,
                                                     (short)0, acc1, false, false);
        acc2 = __builtin_amdgcn_wmma_f32_16x16x4_f32(false, a, false, b2,
                                                     (short)0, acc2, false, false);
    }

    // C/D layout: VGPR v holds row M = v + 8*half, column N = lane&15
    #pragma unroll
    for (int v = 0; v < 8; ++v) {
        const int row = v + 8 * half;
        smat[row][col] = acc1[v];
        if (col == 0) smat[row][16] = acc2[v];
    }
    __syncthreads();

    if (lane < 16) {
        const int frame = F0 + lane;
        float kv[KT];
        float ss = 0.0f;
        #pragma unroll
        for (int k = 0; k < KT; ++k) {
            const float v = smat[lane][k] + ck_b[k];
            kv[k] = v;
            ss += v * v;
        }
        const float inv = 1.0f / (1e-6f + sqrtf(ss));
        float* p = params + (size_t)frame * PSTRIDE;
        #pragma unroll
        for (int k = 0; k < KT; ++k) p[k] = kv[k] * inv;

        const float fg = smat[lane][15] + fg_b[0];
        const float gain = __expf(-fmaxf(fg, 0.0f) + LOG_GAIN_LIMIT);
        const float gg = smat[lane][16] + gg_b[0];
        const float ggain = __expf(GAIN_A * tanhf(gg));   // GAIN_B == 0
        p[KT]     = gain;
        p[KT + 1] = ggain;
    }
}

// ---------------------------------------------------------------------------
// Kernel 2: comb filter + overlap-add crossfade.  One block per (frame, batch).
// Threads 0..199   : current frame's new_chunk[t]
// Threads 200..239 : recompute previous frame's tail (new_chunk_{f-1}[160+..])
// Then crossfade the first OV samples and write FS samples.
// ---------------------------------------------------------------------------
__global__ void __launch_bounds__(256)
comb_synth(const float* __restrict__ x,      // (B, NSAMP)
           const int*   __restrict__ lags,   // (B, NF)
           const float* __restrict__ params, // (B*NF, PSTRIDE)
           float* __restrict__ out)          // (B, NSAMP)
{
    const int f   = blockIdx.x;
    const int b   = blockIdx.y;
    const int tid = threadIdx.x;

    const float* __restrict__ xb = x + (size_t)b * NSAMP;
    const size_t fr = (size_t)b * NF + f;

    __shared__ float pcur[PSTRIDE];
    __shared__ float pprev[PSTRIDE];
    __shared__ float tail[OV];
    __shared__ int   slag[2];

    if (tid < PSTRIDE) {
        pcur[tid] = params[fr * PSTRIDE + tid];
    } else if (tid < 2 * PSTRIDE) {
        const int j = tid - PSTRIDE;
        pprev[j] = (f > 0) ? params[(fr - 1) * PSTRIDE + j] : 0.0f;
    }
    if (tid == 0) {
        slag[0] = lags[fr];
        slag[1] = (f > 0) ? lags[fr - 1] : 0;
    }
    __syncthreads();

    float chunk = 0.0f;
    if (tid < OUTL) {
        // conv[t] = sum_k x[t + k + f*FS - lag - 7] * kern[k]
        const int base = f * FS - slag[0] - 7 + tid;
        float conv = 0.0f;
        #pragma unroll
        for (int k = 0; k < KT; ++k) conv += xat(xb, base + k) * pcur[k];
        const float pt = xat(xb, f * FS + tid);
        chunk = pcur[KT + 1] * (conv * pcur[KT] + pt);
    } else if (tid < OUTL + OV) {
        const int tt = tid - OUTL + FS;   // 160..199 of previous frame
        float tv = 0.0f;
        if (f > 0) {
            const int base = (f - 1) * FS - slag[1] - 7 + tt;
            float conv = 0.0f;
            #pragma unroll
            for (int k = 0; k < KT; ++k) conv += xat(xb, base + k) * pprev[k];
            const float pt = xat(xb, (f - 1) * FS + tt);
            tv = pprev[KT + 1] * (conv * pprev[KT] + pt);
        }
        tail[tt - FS] = tv;
    }
    __syncthreads();

    if (tid < FS) {
        float o = chunk;
        if (tid < OV) {
            const float w2 = 0.5f + 0.5f * __cosf((tid + 0.5f) * PI_F / OV);
            const float w1 = 0.5f + 0.5f * __cosf(((OV - 1 - tid) + 0.5f) * PI_F / OV);
            o = chunk * w1 + tail[tid] * w2;
        }
        out[(size_t)b * NSAMP + f * FS + tid] = o;
    }
}

// ---------------------------------------------------------------------------
extern "C" void kernel_launch(void* const* d_in, const int* in_sizes, int n_in,
                              void* d_out, int out_size, void* d_ws, size_t ws_size,
                              hipStream_t stream) {
    const float* x        = (const float*)d_in[0];  // (B,1,N) f32
    const float* features = (const float*)d_in[1];  // (B,NF,FD) f32
    const int*   lags     = (const int*)  d_in[2];  // (B,NF) i32
    const float* ck_w     = (const float*)d_in[3];  // (K,FD)
    const float* ck_b     = (const float*)d_in[4];  // (K)
    const float* fg_w     = (const float*)d_in[5];  // (1,FD)
    const float* fg_b     = (const float*)d_in[6];  // (1)
    const float* gg_w     = (const float*)d_in[7];  // (1,FD)
    const float* gg_b     = (const float*)d_in[8];  // (1)

    float* out    = (float*)d_out;
    float* params = (float*)d_ws;  // (B*NF) * PSTRIDE floats = 1.28 MB

    const int n_tiles = (NB * NF) / 16;  // 1000
    frame_params_wmma<<<n_tiles, 32, 0, stream>>>(features, ck_w, ck_b,
                                                  fg_w, fg_b, gg_w, gg_b, params);
    comb_synth<<<dim3(NF, NB), 256, 0, stream>>>(x, lags, params, out);
}